// ContrastiveModel_69260642615678
// MI455X (gfx1250) — compile-verified
//
#include <hip/hip_runtime.h>
#include <hip/hip_bf16.h>

typedef __attribute__((ext_vector_type(2))) float v2f;
typedef __attribute__((ext_vector_type(8))) float v8f;

#define B_     8
#define CIN    256
#define HW     16384           // 128*128
#define NPIX   131072          // B*HW
#define DIM    32
#define KQ     2048
#define NCLS   20
#define NSEL   65536           // NPIX/2
#define NLOG   2056            // 1 + (B-1) + KQ
#define INV_T  2.5f
#define MOM    0.999f

// ---- workspace layout (in floats) ----
#define OFF_QN     ((size_t)0)                       // NPIX*DIM
#define OFF_CLMAX  (OFF_QN + (size_t)NPIX*DIM)       // NPIX
#define OFF_QSEL   (OFF_CLMAX + NPIX)                // NSEL*DIM
#define OFF_PACC   (OFF_QSEL + (size_t)NSEL*DIM)     // 256
#define OFF_PROTO  (OFF_PACC + 256)                  // 256
#define OFF_SCAL   (OFF_PROTO + 256)                 // 3: lossPos, lossNeg, nPos
#define OFF_BCNT   (OFF_SCAL + 8)                    // 512 (int)
#define OFF_BOFF   (OFF_BCNT + 512)                  // 512 (int)

__device__ __forceinline__ float waveSum(float v) {
    v += __shfl_xor(v, 16, 32);
    v += __shfl_xor(v, 8, 32);
    v += __shfl_xor(v, 4, 32);
    v += __shfl_xor(v, 2, 32);
    v += __shfl_xor(v, 1, 32);
    return v;
}

// ---------- 0: zero small accumulators ----------
__global__ void k_init(float* __restrict__ ws) {
    for (int i = threadIdx.x; i < 256 + 8; i += blockDim.x)
        ws[OFF_PACC + i] = 0.0f;   // covers PACC(256) then SCAL(8)
}

// ---------- 1: query encoder + bg head + cluster max + BCE partials ----------
__global__ __launch_bounds__(256)
void k_encode_q(const float* __restrict__ im_q, const int* __restrict__ sal_q,
                const float* __restrict__ Wq, const float* __restrict__ Wbg,
                const float* __restrict__ Wcls, float* __restrict__ ws) {
    __shared__ float sW[CIN * DIM + CIN + NCLS * DIM];   // 9088 floats (35.5 KB)
    float* sWq   = sW;
    float* sWbg  = sW + CIN * DIM;
    float* sWcls = sWbg + CIN;
    for (int i = threadIdx.x; i < CIN * DIM; i += 256) sWq[i] = Wq[i];
    for (int i = threadIdx.x; i < CIN;       i += 256) sWbg[i] = Wbg[i];
    for (int i = threadIdx.x; i < NCLS * DIM; i += 256) sWcls[i] = Wcls[i];
    __syncthreads();

    const int p  = blockIdx.x * 256 + threadIdx.x;   // pixel id (grid = 512)
    const int b  = p / HW;
    const int hw = p - b * HW;
    const float* xp = im_q + (size_t)b * CIN * HW + hw;

    float acc[DIM];
#pragma unroll
    for (int d = 0; d < DIM; ++d) acc[d] = 0.0f;
    float abg = 0.0f;

    for (int c = 0; c < CIN; ++c) {
        float x = xp[(size_t)c * HW];
        abg = fmaf(sWbg[c], x, abg);
#pragma unroll
        for (int d = 0; d < DIM; ++d) acc[d] = fmaf(sWq[d * CIN + c], x, acc[d]);
    }

    float s = 0.0f;
#pragma unroll
    for (int d = 0; d < DIM; ++d) s = fmaf(acc[d], acc[d], s);
    float inv = 1.0f / fmaxf(sqrtf(s), 1e-12f);
#pragma unroll
    for (int d = 0; d < DIM; ++d) acc[d] *= inv;

    float* qrow = ws + OFF_QN + (size_t)p * DIM;
#pragma unroll
    for (int d = 0; d < DIM; ++d) qrow[d] = acc[d];

    float cm = -3.4e38f;
#pragma unroll
    for (int o = 0; o < NCLS; ++o) {
        float dt = 0.0f;
#pragma unroll
        for (int d = 0; d < DIM; ++d) dt = fmaf(sWcls[o * DIM + d], acc[d], dt);
        cm = fmaxf(cm, dt);
    }
    ws[OFF_CLMAX + p] = cm;

    // balanced BCE partials (stable log-sigmoid form)
    float lab = (sal_q[p] != 0) ? 1.0f : 0.0f;
    float z   = abg;
    float gz  = (z >= 0.0f) ? 1.0f : 0.0f;
    float lv  = z * (lab - gz) - log1pf(expf(z - 2.0f * z * gz));
    float cp  = -lab * lv;
    float cn  = -(1.0f - lab) * lv;
    cp = waveSum(cp);  cn = waveSum(cn);  float np = waveSum(lab);
    if ((threadIdx.x & 31) == 0) {
        atomicAdd(ws + OFF_SCAL + 0, cp);
        atomicAdd(ws + OFF_SCAL + 1, cn);
        atomicAdd(ws + OFF_SCAL + 2, np);
    }
}

// ---------- 2: momentum key encoder + masked prototype accumulation ----------
__global__ __launch_bounds__(256)
void k_encode_k(const float* __restrict__ im_k, const int* __restrict__ sal_k,
                const float* __restrict__ Wq, const float* __restrict__ Wk,
                float* __restrict__ ws) {
    __shared__ float sWm[CIN * DIM];   // momentum-blended weights (32 KB)
    for (int i = threadIdx.x; i < CIN * DIM; i += 256)
        sWm[i] = MOM * Wk[i] + (1.0f - MOM) * Wq[i];
    __syncthreads();

    const int p  = blockIdx.x * 256 + threadIdx.x;
    const int b  = p / HW;            // uniform per block (HW % 256 == 0)
    const int hw = p - b * HW;
    const float* xp = im_k + (size_t)b * CIN * HW + hw;

    float acc[DIM];
#pragma unroll
    for (int d = 0; d < DIM; ++d) acc[d] = 0.0f;
    for (int c = 0; c < CIN; ++c) {
        float x = xp[(size_t)c * HW];
#pragma unroll
        for (int d = 0; d < DIM; ++d) acc[d] = fmaf(sWm[d * CIN + c], x, acc[d]);
    }
    float s = 0.0f;
#pragma unroll
    for (int d = 0; d < DIM; ++d) s = fmaf(acc[d], acc[d], s);
    float inv = 1.0f / fmaxf(sqrtf(s), 1e-12f);
    float w = (float)sal_k[p];        // salk weight (0/1)

#pragma unroll
    for (int d = 0; d < DIM; ++d) {
        float v = waveSum(acc[d] * inv * w);
        if ((threadIdx.x & 31) == 0) atomicAdd(ws + OFF_PACC + b * DIM + d, v);
    }
}

// ---------- 3: normalize prototypes (one wave per image) ----------
__global__ void k_proto_norm(float* __restrict__ ws) {
    int t = threadIdx.x;              // 256 threads: b = t>>5, d = t&31
    float v = ws[OFF_PACC + t];
    float s = waveSum(v * v);
    ws[OFF_PROTO + t] = v / fmaxf(sqrtf(s), 1e-12f);
}

// ---------- 4: per-block foreground counts ----------
__global__ __launch_bounds__(256)
void k_count(const int* __restrict__ sal_q, float* __restrict__ ws) {
    __shared__ int w8[8];
    int p = blockIdx.x * 256 + threadIdx.x;
    int pred = (sal_q[p] != 0) ? 1 : 0;
    unsigned long long bal = __ballot(pred);
    int lane = threadIdx.x & 31, wave = threadIdx.x >> 5;
    if (lane == 0) w8[wave] = __popcll(bal);
    __syncthreads();
    if (threadIdx.x == 0) {
        int sum = 0;
        for (int i = 0; i < 8; ++i) sum += w8[i];
        ((int*)(ws + OFF_BCNT))[blockIdx.x] = sum;
    }
}

// ---------- 5: single-block exclusive scan over 512 block counts ----------
__global__ __launch_bounds__(512)
void k_scan(float* __restrict__ ws) {
    __shared__ int s[512];
    const int* cnt = (const int*)(ws + OFF_BCNT);
    int* boff = (int*)(ws + OFF_BOFF);
    int t = threadIdx.x;
    int my = cnt[t];
    s[t] = my;
    __syncthreads();
    for (int off = 1; off < 512; off <<= 1) {
        int v = (t >= off) ? s[t - off] : 0;
        __syncthreads();
        s[t] += v;
        __syncthreads();
    }
    boff[t] = s[t] - my;   // exclusive
}

// ---------- 6: compaction + gather q_sel + logits cols 0..7 ----------
__global__ __launch_bounds__(256)
void k_compact(const int* __restrict__ sal_q, float* __restrict__ ws,
               float* __restrict__ out) {
    __shared__ float sP[B_ * DIM];
    __shared__ int w8[8];
    sP[threadIdx.x] = ws[OFF_PROTO + threadIdx.x];   // 256 == B_*DIM
    int p = blockIdx.x * 256 + threadIdx.x;
    int b = p / HW;
    int lane = threadIdx.x & 31, wave = threadIdx.x >> 5;
    int salv = sal_q[p];
    int pred = (salv != 0) ? 1 : 0;
    unsigned long long bal = __ballot(pred);
    int rank = __popcll(bal & ((1ull << lane) - 1ull));
    if (lane == 0) w8[wave] = __popcll(bal);
    __syncthreads();
    int base = 0;
    for (int wv = 0; wv < wave; ++wv) base += w8[wv];

    if (pred) {
        int idx = ((const int*)(ws + OFF_BOFF))[blockIdx.x] + base + rank;
        const float* qr = ws + OFF_QN + (size_t)p * DIM;
        float qv[DIM];
#pragma unroll
        for (int d = 0; d < DIM; ++d) qv[d] = qr[d];
        float* qs = ws + OFF_QSEL + (size_t)idx * DIM;
#pragma unroll
        for (int d = 0; d < DIM; ++d) qs[d] = qv[d];

        int lab  = (salv + 2 * b) * salv;
        int ssel = lab / 2;                       // own image index

        float dots[B_];
#pragma unroll
        for (int j = 0; j < B_; ++j) {
            float dt = 0.0f;
#pragma unroll
            for (int d = 0; d < DIM; ++d) dt = fmaf(sP[j * DIM + d], qv[d], dt);
            dots[j] = dt;
        }
        float* orow = out + (size_t)idx * NLOG;
        orow[0] = ws[OFF_CLMAX + p] * INV_T;
#pragma unroll
        for (int c = 0; c < B_ - 1; ++c) {
            int src = c + ((c >= ssel) ? 1 : 0);  // drop own prototype column
            orow[1 + c] = dots[src] * INV_T;
        }
    }
}

// ---------- 7: bank logits GEMM via V_WMMA_F32_16X16X4_F32 ----------
// q_sel [NSEL x 32] @ obj_queue [32 x 2048] -> out cols [8 .. 2055], scaled 1/T.
// Each wave: 16 rows x 64 cols (4 accumulator tiles), K = 32 -> 8 steps x 4 wmma.
__global__ __launch_bounds__(256)
void k_bank_gemm(const float* __restrict__ obj, const float* __restrict__ ws_ro,
                 float* __restrict__ out) {
    const float* qsel = ws_ro + OFF_QSEL;
    int lane = threadIdx.x & 31, wave = threadIdx.x >> 5;
    int mloc = lane & 15;                 // M for A, N for B, N for C/D
    int k0   = (lane >> 4) * 2;           // K pair base per ISA layout
    int mBase   = blockIdx.y * 128 + wave * 16;
    int colBase = blockIdx.x * 64;

    const float* arow = qsel + (size_t)(mBase + mloc) * DIM;

    v8f acc[4];
#pragma unroll
    for (int t = 0; t < 4; ++t)
#pragma unroll
        for (int r = 0; r < 8; ++r) acc[t][r] = 0.0f;

#pragma unroll
    for (int kk = 0; kk < 8; ++kk) {
        int kb = kk * 4 + k0;             // even -> 8B aligned
        v2f a = *(const v2f*)(arow + kb);
#pragma unroll
        for (int t = 0; t < 4; ++t) {
            const float* bp = obj + (size_t)kb * KQ + colBase + t * 16 + mloc;
            v2f bv;
            bv.x = bp[0];
            bv.y = bp[KQ];
            acc[t] = __builtin_amdgcn_wmma_f32_16x16x4_f32(
                false, a, false, bv, (short)0, acc[t], false, false);
        }
    }

    int rowHi = (lane >> 4) * 8;          // C/D: lanes 16-31 hold rows r+8
#pragma unroll
    for (int t = 0; t < 4; ++t)
#pragma unroll
        for (int r = 0; r < 8; ++r) {
            size_t row = (size_t)(mBase + r + rowHi);
            out[row * NLOG + 8 + colBase + t * 16 + mloc] = acc[t][r] * INV_T;
        }
}

// ---------- 8: pseudo labels, sal_loss, new_queue ----------
__global__ __launch_bounds__(256)
void k_finalize(const float* __restrict__ obj, const float* __restrict__ ws,
                float* __restrict__ out) {
    const size_t L0 = (size_t)NSEL * NLOG;      // pseudo_labels
    const size_t L1 = L0 + NSEL;                // sal_loss
    const size_t L2 = L1 + 1;                   // new_queue [32 x 2048]
    const int total = NSEL + 1 + DIM * KQ;      // 131073
    for (int i = blockIdx.x * 256 + threadIdx.x; i < total; i += 512 * 256) {
        if (i < NSEL) {
            out[L0 + i] = 0.0f;                 // int32 zeros == float 0.0 bits
        } else if (i == NSEL) {
            float lp = ws[OFF_SCAL + 0], ln = ws[OFF_SCAL + 1], np = ws[OFF_SCAL + 2];
            float nn = (float)NPIX - np;
            float tot = (float)NPIX;
            out[L1] = (nn / tot * lp + np / tot * ln) / (float)NPIX;
        } else {
            int j2 = i - NSEL - 1;              // d*2048 + j
            int j  = j2 & (KQ - 1);
            int d  = j2 >> 11;
            out[L2 + j2] = (j < B_) ? ws[OFF_PROTO + j * DIM + d] : obj[j2];
        }
    }
}

extern "C" void kernel_launch(void* const* d_in, const int* in_sizes, int n_in,
                              void* d_out, int out_size, void* d_ws, size_t ws_size,
                              hipStream_t stream) {
    const float* im_q  = (const float*)d_in[0];
    const int*   sal_q = (const int*)d_in[1];
    const float* im_k  = (const float*)d_in[2];
    const int*   sal_k = (const int*)d_in[3];
    const float* Wq    = (const float*)d_in[4];
    const float* Wbg   = (const float*)d_in[5];
    const float* Wk    = (const float*)d_in[6];
    const float* Wcls  = (const float*)d_in[7];
    const float* obj   = (const float*)d_in[8];
    float* out = (float*)d_out;
    float* ws  = (float*)d_ws;
    (void)in_sizes; (void)n_in; (void)out_size; (void)ws_size;

    k_init<<<1, 256, 0, stream>>>(ws);
    k_encode_q<<<NPIX / 256, 256, 0, stream>>>(im_q, sal_q, Wq, Wbg, Wcls, ws);
    k_encode_k<<<NPIX / 256, 256, 0, stream>>>(im_k, sal_k, Wq, Wk, ws);
    k_proto_norm<<<1, 256, 0, stream>>>(ws);
    k_count<<<NPIX / 256, 256, 0, stream>>>(sal_q, ws);
    k_scan<<<1, 512, 0, stream>>>(ws);
    k_compact<<<NPIX / 256, 256, 0, stream>>>(sal_q, ws, out);
    k_bank_gemm<<<dim3(KQ / 64, NSEL / 128), 256, 0, stream>>>(obj, ws, out);
    k_finalize<<<512, 256, 0, stream>>>(obj, ws, out);
}